// ConstantCurrentLIFEncoder_3882650436343
// MI455X (gfx1250) — compile-verified
//
#include <hip/hip_runtime.h>
#include <stdint.h>

// LIF constant-current encoder:
//   v' = v + dt*tau_mem_inv*((v_leak - v) + I) = v + 0.1*(I - v)
//   z  = (v' > v_th)          (v_th = 1.0, strict >)
//   v  = z ? v_reset : v'     (v_reset = 0.0)
// Output: spikes[t, n] for t in [0,100), n in [0, 64*8192)
//
// Roofline: 200 MB float32 spike stream / 23.3 TB/s ~= 8.7 us; VALU is ~100x
// below the roof. No matmul structure (sequential nonlinear recurrence) so
// WMMA does not apply; the job is to saturate the store path.
//
// CDNA5 streaming path: compute spikes into a double-buffered LDS tile
// (ds_store_b128), then drain LDS->global with the async store engine
// (global_store_async_from_lds_b128, ASYNCcnt), 16 B/lane = 512 B/wave per
// step, non-temporal (output > 192 MB L2, never re-read). Double buffering +
// s_wait_asynccnt 1 keeps two async stores in flight per wave while the VALU
// computes the next step.

typedef float v4f __attribute__((ext_vector_type(4)));

#define LIF_SEQ 100

__global__ __launch_bounds__(256) void lif_scan_kernel(const float* __restrict__ in,
                                                       float* __restrict__ out,
                                                       int n) {
    __shared__ v4f buf[2][256];  // 8 KB: double-buffered spike tile (16 B/thread)

    const int tid = threadIdx.x;
    const int i = (blockIdx.x * blockDim.x + tid) * 4;
    if (i >= n) return;

    // One float4 of constant input current per thread (read once, coalesced).
    const v4f I = *reinterpret_cast<const v4f*>(in + i);

    float v0 = 0.0f, v1 = 0.0f, v2 = 0.0f, v3 = 0.0f;

    // Per-thread global byte address of out[0, i]; advances by one time-step
    // row (n floats) per iteration.
    uint64_t gaddr = (uint64_t)(uintptr_t)(out + i);
    const uint64_t gstride = (uint64_t)n * 4u;

    // LDS byte offsets of this thread's two parity slots (low 32 bits of the
    // generic pointer are the LDS offset in the LDS aperture).
    const uint32_t loff0 = (uint32_t)(uintptr_t)&buf[0][tid];
    const uint32_t loff1 = (uint32_t)(uintptr_t)&buf[1][tid];

    #pragma unroll 2
    for (int t = 0; t < LIF_SEQ; ++t) {
        // integrate: v += 0.1 * (I - v)
        v0 += 0.1f * (I.x - v0);
        v1 += 0.1f * (I.y - v1);
        v2 += 0.1f * (I.z - v2);
        v3 += 0.1f * (I.w - v3);

        // spike (strict >) and reset to 0
        const float z0 = (v0 > 1.0f) ? 1.0f : 0.0f;
        const float z1 = (v1 > 1.0f) ? 1.0f : 0.0f;
        const float z2 = (v2 > 1.0f) ? 1.0f : 0.0f;
        const float z3 = (v3 > 1.0f) ? 1.0f : 0.0f;
        v0 = (v0 > 1.0f) ? 0.0f : v0;
        v1 = (v1 > 1.0f) ? 0.0f : v1;
        v2 = (v2 > 1.0f) ? 0.0f : v2;
        v3 = (v3 > 1.0f) ? 0.0f : v3;

        v4f z = { z0, z1, z2, z3 };

        // Reuse of this parity slot: the async store issued 2 iterations ago
        // must have read it out of LDS. Keep at most 2 in flight (<=1 older).
        asm volatile("s_wait_asynccnt 0x1" ::: "memory");

        // Stage spike tile in LDS (ds_store_b128, 16 B aligned).
        buf[t & 1][tid] = z;

        // Async engine reads LDS; DS write must be complete first.
        asm volatile("s_wait_dscnt 0x0" ::: "memory");

        // Drain LDS -> global: 16 B/lane coalesced, non-temporal (streamed
        // 200 MB output, never re-read, larger than L2).
        const uint32_t loff = (t & 1) ? loff1 : loff0;
        asm volatile("global_store_async_from_lds_b128 %0, %1, off th:TH_STORE_NT"
                     :: "v"(gaddr), "v"(loff)
                     : "memory");

        gaddr += gstride;
    }

    // Ensure all async LDS->global traffic is complete before wave exit.
    asm volatile("s_wait_asynccnt 0x0" ::: "memory");
}

extern "C" void kernel_launch(void* const* d_in, const int* in_sizes, int n_in,
                              void* d_out, int out_size, void* d_ws, size_t ws_size,
                              hipStream_t stream) {
    (void)n_in; (void)out_size; (void)d_ws; (void)ws_size;

    const float* in = reinterpret_cast<const float*>(d_in[0]);
    float* out = reinterpret_cast<float*>(d_out);
    const int n = in_sizes[0];            // 64 * 8192 = 524288 neurons

    const int threads = 256;              // 8 wave32s per block
    const int vec_elems = 4;              // float4 per thread
    const int n_threads = (n + vec_elems - 1) / vec_elems;
    const int blocks = (n_threads + threads - 1) / threads;   // 512 blocks

    lif_scan_kernel<<<blocks, threads, 0, stream>>>(in, out, n);
}